// VoxelMapEncoder_56212531970409
// MI455X (gfx1250) — compile-verified
//
#include <hip/hip_runtime.h>

// ---------------- problem constants (match reference) ----------------
#define IMG   224
#define NXV   480
#define NYV   480
#define NZV   56
#define NB    4
#define MIN_D 0.1f
#define MAX_D 1.5f
#define INV_VOXEL 20.0f            // 1 / 0.05, guess only; exactness comes from fixup

#define NEX (NXV + 1)              // 481 edges
#define NEY (NYV + 1)              // 481 edges
#define NEZ (NZV + 1)              // 57 edges
#define NBINS (NEX + NEY + NEZ)    // 1019 floats in LDS

typedef float v2f __attribute__((ext_vector_type(2)));
typedef float v4f __attribute__((ext_vector_type(4)));
typedef float v8f __attribute__((ext_vector_type(8)));

// ---------------- kernel 1: zero the 206MB voxel map -----------------
// Store-bandwidth bound (~9us @ 23.3TB/s): b128 non-temporal stores.
__global__ __launch_bounds__(256)
void vox_clear_kernel(v4f* __restrict__ out, int n4) {
    int i = blockIdx.x * 256 + threadIdx.x;
    if (i < n4) {
        v4f z = {0.f, 0.f, 0.f, 0.f};
        __builtin_nontemporal_store(z, &out[i]);   // global_store_b128 th:TH_NT
    }
}

// searchsorted(bins, x, 'right') - 1, exact for monotone bins:
// uniform-grid initial guess, then bounded fixup against the real edges (LDS).
__device__ __forceinline__ int bucket(const float* __restrict__ bins, int n /*cells*/, float x) {
    int i = (int)floorf(x * INV_VOXEL);
    i = min(max(i, 0), n - 1);
    while (i < n && bins[i + 1] <= x) ++i;   // typically 0-1 iterations
    while (i >= 0 && bins[i] > x)     --i;   // typically 0-1 iterations
    return i;                                 // in [-1, n]; valid iff [0, n)
}

// ---------------- kernel 2: rotate rays (WMMA) + scatter -------------
// Per 32-lane wave: 32 rays via two V_WMMA_F32_16X16X4_F32.
//   WMMA#1: A rows 0..2  = rot, B cols = rays [base .. base+15]
//   WMMA#2: A rows 8..10 = rot, B cols = rays [base+16 .. base+31]
// D layout (16x16 f32): lanes 0-15 hold rows 0-7, lanes 16-31 hold rows 8-15,
// so lane L ends up with world dir of ray (base+L) in D[0..2] - no shuffles.
__global__ __launch_bounds__(256)
void vox_scatter_kernel(const float* __restrict__ rays,   // [H,W,3]
                        const float* __restrict__ bx,     // [NX+1]
                        const float* __restrict__ by,     // [NY+1]
                        const float* __restrict__ bz,     // [NZ+1]
                        const float* __restrict__ sb,     // [B,3]
                        const float* __restrict__ feat,   // [B,H,W]
                        const float* __restrict__ dep,    // [B,H,W]
                        const float* __restrict__ pos,    // [B,3]
                        const float* __restrict__ at,     // [B,3]
                        const float* __restrict__ up,     // [B,3]
                        float* __restrict__ out)          // [B,NX,NY,NZ]
{
    __shared__ float sbins[NBINS];   // [ex(481) | ey(481) | ez(57)]

    const int tid = threadIdx.x;

    // cooperative LDS fill of all bin edges (coalesced, 4 iters/thread)
    for (int i = tid; i < NBINS; i += 256) {
        float v;
        if (i < NEX)            v = bx[i];
        else if (i < NEX + NEY) v = by[i - NEX];
        else                    v = bz[i - NEX - NEY];
        sbins[i] = v;
    }
    __syncthreads();                 // EXEC back to all-1s before WMMA

    const int lane  = tid & 31;
    const int gbase = blockIdx.x * 256 + (tid & ~31);     // first global ray of wave
    const int HW    = IMG * IMG;
    const int b     = gbase / HW;                         // uniform (50176 % 256 == 0)
    const int ray0  = gbase - b * HW;

    // camera frame: eye = at, u = up, c = cross(at, up)
    const float ax = at[b*3+0], ay = at[b*3+1], az = at[b*3+2];
    const float ux = up[b*3+0], uy = up[b*3+1], uz = up[b*3+2];
    const float cx = ay*uz - az*uy;
    const float cy = az*ux - ax*uz;
    const float cz = ax*uy - ay*ux;
    const float ox = pos[b*3+0] - sb[b*3+0];
    const float oy = pos[b*3+1] - sb[b*3+1];
    const float oz = pos[b*3+2] - sb[b*3+2];

    // rot row m = [c[m], u[m], -e[m], 0]
    const int  m    = lane & 15;     // A/B row/col index held by this lane
    const bool lo16 = lane < 16;     // lanes 0-15: K=0,1 ; lanes 16-31: K=2,3

    const float r1k0 = (m==0)?cx : (m==1)?cy : (m==2)?cz : 0.f;
    const float r1k1 = (m==0)?ux : (m==1)?uy : (m==2)?uz : 0.f;
    const float r1k2 = (m==0)?-ax: (m==1)?-ay: (m==2)?-az: 0.f;
    v2f a1; a1.x = lo16 ? r1k0 : r1k2;  a1.y = lo16 ? r1k1 : 0.f;

    const float r2k0 = (m==8)?cx : (m==9)?cy : (m==10)?cz : 0.f;
    const float r2k1 = (m==8)?ux : (m==9)?uy : (m==10)?uz : 0.f;
    const float r2k2 = (m==8)?-ax: (m==9)?-ay: (m==10)?-az: 0.f;
    v2f a2; a2.x = lo16 ? r2k0 : r2k2;  a2.y = lo16 ? r2k1 : 0.f;

    const int p1 = ray0 + m;
    const int p2 = p1 + 16;
    const float r1x = rays[p1*3+0], r1y = rays[p1*3+1], r1z = rays[p1*3+2];
    const float r2x = rays[p2*3+0], r2y = rays[p2*3+1], r2z = rays[p2*3+2];
    v2f b1; b1.x = lo16 ? r1x : r1z;  b1.y = lo16 ? r1y : 0.f;
    v2f b2; b2.x = lo16 ? r2x : r2z;  b2.y = lo16 ? r2y : 0.f;

    v8f czero = {};
    v8f d1 = __builtin_amdgcn_wmma_f32_16x16x4_f32(false, a1, false, b1,
                                                   (short)0, czero, false, false);
    v8f d2 = __builtin_amdgcn_wmma_f32_16x16x4_f32(false, a2, false, b2,
                                                   (short)0, czero, false, false);

    const float wx = lo16 ? d1[0] : d2[0];
    const float wy = lo16 ? d1[1] : d2[1];
    const float wz = lo16 ? d1[2] : d2[2];

    // this lane owns ray (gbase + lane)
    const int pix = ray0 + lane;
    const float d = dep [b*HW + pix];
    const float f = feat[b*HW + pix];

    const float px = ox + wx * d;
    const float py = oy + wy * d;
    const float pz = oz + wz * d;

    const int ix = bucket(sbins,             NXV, px);
    const int iy = bucket(sbins + NEX,       NYV, py);
    const int iz = bucket(sbins + NEX + NEY, NZV, pz);

    const bool valid = (d >= MIN_D) & (d <= MAX_D)
                     & (ix >= 0) & (ix < NXV)
                     & (iy >= 0) & (iy < NYV)
                     & (iz >= 0) & (iz < NZV);
    if (valid) {
        const int flat = ((b * NXV + ix) * NYV + iy) * NZV + iz;
        out[flat] = f;   // .set semantics: plain store, arbitrary winner on collision
    }
}

// ---------------------------- launcher -------------------------------
extern "C" void kernel_launch(void* const* d_in, const int* in_sizes, int n_in,
                              void* d_out, int out_size, void* d_ws, size_t ws_size,
                              hipStream_t stream) {
    const float* rays = (const float*)d_in[0];
    const float* bx   = (const float*)d_in[1];
    const float* by   = (const float*)d_in[2];
    const float* bz   = (const float*)d_in[3];
    const float* sb   = (const float*)d_in[4];
    const float* feat = (const float*)d_in[5];
    const float* dep  = (const float*)d_in[6];
    const float* pos  = (const float*)d_in[7];
    const float* at   = (const float*)d_in[8];
    const float* up   = (const float*)d_in[9];
    float* out = (float*)d_out;

    // 1) zero the voxel map: out_size = 4*480*480*56 (divisible by 4)
    const int n4 = out_size / 4;
    const int cblocks = (n4 + 255) / 256;
    vox_clear_kernel<<<cblocks, 256, 0, stream>>>((v4f*)out, n4);

    // 2) rotate + scatter: 4*224*224 = 200704 rays, 256 rays/block
    const int nrays = NB * IMG * IMG;
    vox_scatter_kernel<<<nrays / 256, 256, 0, stream>>>(
        rays, bx, by, bz, sb, feat, dep, pos, at, up, out);
}